// SmoothSkeletonDual_4398046511110
// MI455X (gfx1250) — compile-verified
//
#include <hip/hip_runtime.h>
#include <stdint.h>

// Problem constants (from the reference)
#define NB 2
#define NH 1024
#define NW 1024
#define TILE 64
#define NITER 10

// alpha = 0.05
#define EXP_SCALE 28.853900817779268f   // (1/alpha) * log2(e)
#define A_LN2     0.034657359027997264f // alpha * ln(2)
#define A_LOG9    0.10986122886681098f  // alpha * log(9)

// LDS stage extents (iteration kernel)
#define H3 (TILE + 6)   // 70 : xc tile + halo 3
#define H2 (TILE + 4)   // 68 : y  = erode(xc)
#define H1 (TILE + 2)   // 66 : z  = erode(y)
// init kernel
#define H2i (TILE + 4)  // 68 : x tile + halo 2
#define H1i (TILE + 2)  // 66 : e = erode(x)

__device__ __forceinline__ int reflect_i(int q, int n) {
    q = (q < 0) ? -q : q;
    q = (q >= n) ? (2 * n - 2 - q) : q;
    return q;
}

// Soft morphology over a 3x3 neighborhood in LDS.
// sgn = +1 -> soft dilate, sgn = -1 -> soft erode.
//   u = sgn*v ;  out = sgn*( m + alpha*ln2*log2( sum 2^((u-m)*EXP_SCALE) ) - alpha*log 9 )
// == softmax-weighted-sum + alpha*(entropy - logN) up to the reference's EPS (~1e-6 abs).
// v_exp_f32 / v_log_f32 are natively base-2 on AMDGPU.
__device__ __forceinline__ float smorph9(const float* base, int stride, float sgn) {
    float u[9];
#pragma unroll
    for (int a = 0; a < 3; ++a)
#pragma unroll
        for (int c = 0; c < 3; ++c)
            u[a * 3 + c] = sgn * base[a * stride + c];
    float m = u[0];
#pragma unroll
    for (int k = 1; k < 9; ++k) m = fmaxf(m, u[k]);
    float s = 0.f;
#pragma unroll
    for (int k = 0; k < 9; ++k)
        s += __builtin_amdgcn_exp2f((u[k] - m) * EXP_SCALE);   // v_exp_f32
    return sgn * (m + A_LN2 * __builtin_amdgcn_logf(s) - A_LOG9);  // v_log_f32
}

// Async global->LDS fill of one pre-reflected tile (CDNA5 async path, ASYNCcnt).
__device__ __forceinline__ void async_fill_tile(const float* __restrict__ plane,
                                                float* lds, int ext, int halo,
                                                int ty, int tx, int tid) {
    const int n = ext * ext;
    for (int s = tid; s < n; s += 256) {
        const int i = s / ext, j = s - i * ext;
        const int gr = reflect_i(ty + i - halo, NH);
        const int gc = reflect_i(tx + j - halo, NW);
        const float* gp = plane + (size_t)gr * NW + gc;
        const uint32_t lp = (uint32_t)(uintptr_t)(&lds[s]);
        asm volatile("global_load_async_to_lds_b32 %0, %1, off"
                     :: "v"(lp), "v"(gp) : "memory");
    }
    asm volatile("s_wait_asynccnt 0x0" ::: "memory");
    __syncthreads();
}

// ---------------- init: skel = clip(x - dilate(erode(x)), 0, 1) ----------------
__global__ __launch_bounds__(256)
void skel_init_kernel(const float* __restrict__ x, float* __restrict__ skel) {
    __shared__ float s0[H2i * H2i];   // x, pre-reflected, halo 2
    __shared__ float s1[H1i * H1i];   // e = erode(x), halo 1

    const int tid = threadIdx.x;
    const int tx = blockIdx.x * TILE, ty = blockIdx.y * TILE;
    const size_t pbase = (size_t)blockIdx.z * NH * NW;
    const float* plane = x + pbase;

    async_fill_tile(plane, s0, H2i, 2, ty, tx, tid);

    for (int s = tid; s < H1i * H1i; s += 256) {
        const int i = s / H1i, j = s - i * H1i;
        s1[s] = smorph9(&s0[i * H2i + j], H2i, -1.0f);   // erode
    }
    __syncthreads();

    for (int s = tid; s < TILE * TILE; s += 256) {
        const int i = s >> 6, j = s & 63;
        const float o  = smorph9(&s1[i * H1i + j], H1i, +1.0f);  // dilate(e) = open(x)
        const float xc = s0[(i + 2) * H2i + (j + 2)];
        float d = xc - o;
        d = fminf(fmaxf(d, 0.f), 1.f);
        skel[pbase + (size_t)(ty + i) * NW + (tx + j)] = d;      // initialize d_out
    }
}

// ------- one skeleton iteration, fully fused (3 stencil passes in one launch) -------
// y = erode(src); z = erode(y); w = dilate(z); skel += clip(y - w, 0, 1); ydst = y
__global__ __launch_bounds__(256)
void skel_iter_kernel(const float* __restrict__ src, float* __restrict__ ydst,
                      float* __restrict__ skel) {
    __shared__ float s0[H3 * H3];   // xc, pre-reflected, halo 3
    __shared__ float s1[H2 * H2];   // y,  halo 2
    __shared__ float s2[H1 * H1];   // z,  halo 1

    const int tid = threadIdx.x;
    const int tx = blockIdx.x * TILE, ty = blockIdx.y * TILE;
    const size_t pbase = (size_t)blockIdx.z * NH * NW;
    const float* plane = src + pbase;

    async_fill_tile(plane, s0, H3, 3, ty, tx, tid);

    for (int s = tid; s < H2 * H2; s += 256) {
        const int i = s / H2, j = s - i * H2;
        s1[s] = smorph9(&s0[i * H3 + j], H3, -1.0f);     // y = erode(xc)
    }
    __syncthreads();

    for (int s = tid; s < H1 * H1; s += 256) {
        const int i = s / H1, j = s - i * H1;
        s2[s] = smorph9(&s1[i * H2 + j], H2, -1.0f);     // z = erode(y)
    }
    __syncthreads();

    for (int s = tid; s < TILE * TILE; s += 256) {
        const int i = s >> 6, j = s & 63;
        const float w  = smorph9(&s2[i * H1 + j], H1, +1.0f);   // w = dilate(z) = open(y)
        const float yc = s1[(i + 2) * H2 + (j + 2)];
        const size_t g = pbase + (size_t)(ty + i) * NW + (tx + j);
        ydst[g] = yc;                                    // next xc
        float d = yc - w;
        d = fminf(fmaxf(d, 0.f), 1.f);
        skel[g] += d;                                    // accumulate skeleton
    }
}

extern "C" void kernel_launch(void* const* d_in, const int* in_sizes, int n_in,
                              void* d_out, int out_size, void* d_ws, size_t ws_size,
                              hipStream_t stream) {
    (void)in_sizes; (void)n_in; (void)out_size; (void)ws_size;
    const float* x = (const float*)d_in[0];
    float* skel = (float*)d_out;
    float* buf0 = (float*)d_ws;
    float* buf1 = buf0 + (size_t)NB * NH * NW;   // two 8MB ping-pong planes in workspace

    dim3 grid(NW / TILE, NH / TILE, NB);
    dim3 block(256);

    skel_init_kernel<<<grid, block, 0, stream>>>(x, skel);

    const float* src = x;
    for (int it = 0; it < NITER; ++it) {
        float* dst = (it & 1) ? buf1 : buf0;
        skel_iter_kernel<<<grid, block, 0, stream>>>(src, dst, skel);
        src = dst;
    }
}